// GCNConv_66907000537770
// MI455X (gfx1250) — compile-verified
//
#include <hip/hip_runtime.h>

#define DIM 64
#define LDA 68  // padded LDS row stride in floats: bank = (4m + k) % 64 -> conflict-free

typedef float v2f __attribute__((ext_vector_type(2)));
typedef float v8f __attribute__((ext_vector_type(8)));

// Block = 128 threads = 4 waves. Each wave owns one 16-wide column tile of the
// DIM=64 output; the 16x64 A tile is staged once in LDS and shared by all 4
// waves and both GEMMs (C0 = A@B0, C1 = A@B1). Full-f32 WMMA (16x16x4).
__global__ __launch_bounds__(128)
void gcn_gemm2_wmma_f32(const float* __restrict__ A,
                        const float* __restrict__ B0,
                        const float* __restrict__ B1,
                        float* __restrict__ C0,
                        float* __restrict__ C1,
                        int N) {
    __shared__ float tileA[16 * LDA];
    const int rowBase = blockIdx.x * 16;
    const int t = threadIdx.x;

    // Cooperative load of the 16x64 A tile: 8 consecutive floats per thread.
    {
        const int r  = t >> 3;        // 0..15
        const int cb = (t & 7) * 8;   // 0..56
        float4 u0 = make_float4(0.f, 0.f, 0.f, 0.f);
        float4 u1 = make_float4(0.f, 0.f, 0.f, 0.f);
        if (rowBase + r < N) {
            const float4* src = (const float4*)(A + (size_t)(rowBase + r) * DIM + cb);
            u0 = src[0];
            u1 = src[1];
        }
        float* dst = &tileA[r * LDA + cb];            // 8-byte aligned (cb even)
        ((float2*)dst)[0] = make_float2(u0.x, u0.y);
        ((float2*)dst)[1] = make_float2(u0.z, u0.w);
        ((float2*)dst)[2] = make_float2(u1.x, u1.y);
        ((float2*)dst)[3] = make_float2(u1.z, u1.w);
    }
    __syncthreads();

    const int wid  = t >> 5;        // wave id -> 16-wide column tile
    const int lane = t & 31;
    const int half = lane >> 4;     // lane half selects K pair {0,1} vs {2,3}
    const int m    = lane & 15;     // A row / B,C column within the 16x16 tile
    const int nCol = wid * 16 + m;  // global output column (0..63)
    const int koff = half * 2;

    v8f acc0 = {};
    v8f acc1 = {};
#pragma unroll
    for (int kk = 0; kk < DIM; kk += 4) {
        v2f a, b0, b1;
        // A 16x4 fragment (ISA layout): VGPR0 -> K = koff, VGPR1 -> K = koff+1
        a.x = tileA[m * LDA + kk + koff];
        a.y = tileA[m * LDA + kk + koff + 1];
        // B 4x16 fragment: row K striped across lanes, halves split rows
        b0.x = B0[(size_t)(kk + koff) * DIM + nCol];
        b0.y = B0[(size_t)(kk + koff + 1) * DIM + nCol];
        b1.x = B1[(size_t)(kk + koff) * DIM + nCol];
        b1.y = B1[(size_t)(kk + koff + 1) * DIM + nCol];
        acc0 = __builtin_amdgcn_wmma_f32_16x16x4_f32(false, a, false, b0,
                                                     (short)0, acc0, false, false);
        acc1 = __builtin_amdgcn_wmma_f32_16x16x4_f32(false, a, false, b1,
                                                     (short)0, acc1, false, false);
    }

    // C/D layout: VGPR r -> M = r (lanes 0-15) / M = r+8 (lanes 16-31), N = lane%16
#pragma unroll
    for (int r = 0; r < 8; ++r) {
        const int gRow = rowBase + half * 8 + r;
        if (gRow < N) {
            C0[(size_t)gRow * DIM + nCol] = acc0[r];
            C1[(size_t)gRow * DIM + nCol] = acc1[r];
        }
    }
}

// row_ptr[i] = lower_bound(row_ids, i), i in [0, N]  (row_ids is sorted)
__global__ __launch_bounds__(256)
void gcn_row_ptr(const int* __restrict__ row_ids, int E, int N,
                 int* __restrict__ rp) {
    const int i = blockIdx.x * blockDim.x + threadIdx.x;
    if (i > N) return;
    int lo = 0, hi = E;
    while (lo < hi) {
        const int mid = (lo + hi) >> 1;
        if (row_ids[mid] < i) lo = mid + 1; else hi = mid;
    }
    rp[i] = lo;
}

// One wave per node: deterministic sequential edge sum, 256B coalesced gathers
// (32 lanes x float2 spans one full XW row). out already holds X@W1.
__global__ __launch_bounds__(256)
void gcn_spmm_agg(const float* __restrict__ XW,
                  const int* __restrict__ rp,
                  const int* __restrict__ cols,
                  const float* __restrict__ dd1,
                  float* __restrict__ out, int N) {
    const int lane = threadIdx.x & 31;
    const int node = blockIdx.x * 8 + (threadIdx.x >> 5);
    if (node >= N) return;

    const int lo = rp[node];
    const int hi = rp[node + 1];

    float ax = 0.f, ay = 0.f;
    const float2* xw2 = (const float2*)XW;
    for (int e = lo; e < hi; ++e) {
        const int c = cols[e];
        if (e + 1 < hi) {
            // global_prefetch_b8 of the next gathered row
            __builtin_prefetch(XW + (size_t)cols[e + 1] * DIM, 0, 1);
        }
        const float2 v = xw2[(size_t)c * (DIM / 2) + lane];
        ax += v.x;
        ay += v.y;
    }

    const float d = dd1[node];
    float2* o = (float2*)out + (size_t)node * (DIM / 2) + lane;
    float2 prev = *o;   // X@W1 contribution written by the GEMM kernel
    prev.x += ax / d;
    prev.y += ay / d;
    *o = prev;
}

extern "C" void kernel_launch(void* const* d_in, const int* in_sizes, int n_in,
                              void* d_out, int out_size, void* d_ws, size_t ws_size,
                              hipStream_t stream) {
    const float* X       = (const float*)d_in[0];
    const float* W       = (const float*)d_in[1];
    const float* W1      = (const float*)d_in[2];
    const float* dd1     = (const float*)d_in[3];
    const int*   row_ids = (const int*)d_in[4];
    const int*   col_ids = (const int*)d_in[5];

    const int N = in_sizes[0] / DIM;   // 100000
    const int E = in_sizes[4];         // 1600000

    float* out = (float*)d_out;
    float* XW  = (float*)d_ws;                                   // N*64 floats
    int*   rp  = (int*)((char*)d_ws + (size_t)N * DIM * sizeof(float)); // N+1 ints

    // 1) XW = X@W (to workspace) and out = X@W1, fused WMMA GEMMs
    const int tiles = (N + 15) / 16;
    gcn_gemm2_wmma_f32<<<tiles, 128, 0, stream>>>(X, W, W1, XW, out, N);

    // 2) CSR row pointers from sorted row_ids
    gcn_row_ptr<<<(N + 1 + 255) / 256, 256, 0, stream>>>(row_ids, E, N, rp);

    // 3) out = segment_sum(XW[col]) / dd1 + out
    gcn_spmm_agg<<<(N + 7) / 8, 256, 0, stream>>>(XW, rp, col_ids, dd1, out, N);
}